// MyModel_87522843561376
// MI455X (gfx1250) — compile-verified
//
#include <hip/hip_runtime.h>
#include <hip/hip_bf16.h>

typedef __attribute__((ext_vector_type(2))) float v2f;
typedef __attribute__((ext_vector_type(8))) float v8f;

__device__ __forceinline__ float rcp_fast(float v) {
    // hardware v_rcp_f32 (~1 ulp), avoids the ~13-op IEEE div expansion
    return __builtin_amdgcn_rcpf(v);
}

__device__ __forceinline__ float sigmoid_f(float v) {
    // v_exp_f32 + v_add + v_rcp_f32, fully straight-line
    return rcp_fast(1.0f + __expf(-v));
}

__device__ __forceinline__ float tanh_fast(float v) {
    // tanh(x) = 1 - 2/(exp(2x)+1); branch-free, saturates to +/-1 at the
    // extremes (exp -> inf gives 1, exp -> 0 gives -1).
    return 1.0f - 2.0f * rcp_fast(__expf(2.0f * v) + 1.0f);
}

// Single-step LSTM (B=1, T=1, in=1, units=1), whole step in one wave32.
// z = [x, h] @ [kernel; recurrent_kernel] + bias  computed with one
// V_WMMA_F32_16X16X4_F32 (M=16, N=16, K=4 — live region is M=1,K=2,N=4).
__global__ void lstm1_step_wmma(const float* __restrict__ x,
                                const float* __restrict__ h0,
                                const float* __restrict__ c0,
                                const float* __restrict__ W,    // kernel (1,4)
                                const float* __restrict__ U,    // recurrent_kernel (1,4)
                                const float* __restrict__ bias, // (4,)
                                float* __restrict__ out) {
    const int lane = threadIdx.x;   // 0..31 (wave32)

    const float xv = x[0];
    const float hv = h0[0];
    const float cv = c0[0];

    // A-matrix 16x4 f32: lanes 0-15 hold rows M=0..15 with a[0]=K0, a[1]=K1;
    // lanes 16-31 hold K2/K3. Live element: A[0][0]=x, A[0][1]=h (lane 0).
    v2f a;
    a.x = (lane == 0) ? xv : 0.0f;
    a.y = (lane == 0) ? hv : 0.0f;

    // B-matrix 4x16 f32: rows striped across lanes within a VGPR.
    // b[0]: lanes 0-15 = row K=0 (N=lane), lanes 16-31 = row K=2
    // b[1]: lanes 0-15 = row K=1,          lanes 16-31 = row K=3
    v2f b;
    b.x = (lane < 4) ? W[lane] : 0.0f;   // B[0][n] = kernel[0][n]
    b.y = (lane < 4) ? U[lane] : 0.0f;   // B[1][n] = recurrent_kernel[0][n]

    // C-matrix 16x16 f32: c[0] lane n = C[0][n] = bias[n].
    v8f c = {};
    c[0] = (lane < 4) ? bias[lane] : 0.0f;

    // D = A*B + C  -> d[0] at lane n holds z_n = x*W[n] + h*U[n] + bias[n]
    v8f d = __builtin_amdgcn_wmma_f32_16x16x4_f32(
        /*neg_a=*/false, a, /*neg_b=*/false, b,
        /*c_mod=*/(short)0, c, /*reuse_a=*/false, /*reuse_b=*/false);

    const float zrow = d[0];
    const float zi = __shfl(zrow, 0, 32);
    const float zf = __shfl(zrow, 1, 32);
    const float zc = __shfl(zrow, 2, 32);
    const float zo = __shfl(zrow, 3, 32);

    const float ig = sigmoid_f(zi);
    const float fg = sigmoid_f(zf);
    const float cn = fg * cv + ig * tanh_fast(zc);
    const float og = sigmoid_f(zo);
    const float hn = og * tanh_fast(cn);

    if (lane == 0) out[0] = hn;
}

extern "C" void kernel_launch(void* const* d_in, const int* in_sizes, int n_in,
                              void* d_out, int out_size, void* d_ws, size_t ws_size,
                              hipStream_t stream) {
    (void)in_sizes; (void)n_in; (void)out_size; (void)d_ws; (void)ws_size;
    const float* x    = (const float*)d_in[0]; // (1,1,1)
    const float* h0   = (const float*)d_in[1]; // (1,1)
    const float* c0   = (const float*)d_in[2]; // (1,1)
    const float* W    = (const float*)d_in[3]; // (1,4)
    const float* U    = (const float*)d_in[4]; // (1,4)
    const float* bias = (const float*)d_in[5]; // (4,)
    float* out = (float*)d_out;                // (1,1)

    lstm1_step_wmma<<<1, 32, 0, stream>>>(x, h0, c0, W, U, bias, out);
}